// CustomKilLayer_6244882448616
// MI455X (gfx1250) — compile-verified
//
#include <hip/hip_runtime.h>
#include <math.h>

#define NN 4096
#define DD 256
#define RR 8
#define LN_EPS 1e-5f

typedef __attribute__((ext_vector_type(2))) float v2f;
typedef __attribute__((ext_vector_type(8))) float v8f;

// ---------------------------------------------------------------------------
// K1: relation_pred collapsed to the single row that matters.
//   row = token_index[node_index]
//   q = inputs_embeds[row] @ W_q + b_q ; qn = LN(q) ; p = softmax(qn @ rels)
//   w[r] = wrel[r] * p[r]
// One block of 256 threads (8 waves); thread t owns output column t.
// ---------------------------------------------------------------------------
__global__ __launch_bounds__(256) void k_relpred(
    const float* __restrict__ inputs_embeds,
    const int*   __restrict__ token_index,
    const int*   __restrict__ node_index,
    const float* __restrict__ rels,
    const float* __restrict__ wrel,
    const float* __restrict__ W_q,
    const float* __restrict__ b_q,
    float*       __restrict__ w_out)
{
    const int tid  = threadIdx.x;
    const int node = node_index[0];
    const int row  = token_index[node];
    const float* x = inputs_embeds + (size_t)row * DD;

    // q[tid] = sum_k x[k] * W_q[k, tid] + b_q[tid]
    float acc = b_q[tid];
    #pragma unroll 8
    for (int k = 0; k < DD; ++k)
        acc = fmaf(x[k], W_q[k * DD + tid], acc);

    __shared__ float red[DD];
    // mean
    red[tid] = acc;
    __syncthreads();
    for (int st = 128; st > 0; st >>= 1) {
        if (tid < st) red[tid] += red[tid + st];
        __syncthreads();
    }
    const float mean = red[0] * (1.0f / DD);
    __syncthreads();
    // variance
    const float dm = acc - mean;
    red[tid] = dm * dm;
    __syncthreads();
    for (int st = 128; st > 0; st >>= 1) {
        if (tid < st) red[tid] += red[tid + st];
        __syncthreads();
    }
    const float var = red[0] * (1.0f / DD);
    const float qn  = dm * rsqrtf(var + LN_EPS);

    // logits[r] = sum_col qn[col] * rels[col, r]
    __shared__ float logits[RR];
    if (tid < RR) logits[tid] = 0.0f;
    __syncthreads();
    #pragma unroll
    for (int r = 0; r < RR; ++r)
        atomicAdd(&logits[r], qn * rels[tid * RR + r]);
    __syncthreads();

    if (tid == 0) {
        float mx = logits[0];
        #pragma unroll
        for (int r = 1; r < RR; ++r) mx = fmaxf(mx, logits[r]);
        float ex[RR], se = 0.0f;
        #pragma unroll
        for (int r = 0; r < RR; ++r) { ex[r] = __expf(logits[r] - mx); se += ex[r]; }
        const float inv = 1.0f / se;
        #pragma unroll
        for (int r = 0; r < RR; ++r) w_out[r] = wrel[r] * ex[r] * inv;
    }
}

// ---------------------------------------------------------------------------
// K2: the bandwidth kernel. For each (r, i): rowsum of A[r][i][:] and A[r][i][i].
// 536 MB streamed once; float4 loads, 256 threads per row.
// ---------------------------------------------------------------------------
__global__ __launch_bounds__(256) void k_rowsum(
    const float* __restrict__ A,
    float*       __restrict__ S,
    float*       __restrict__ Dg)
{
    const int i = blockIdx.x;
    const int r = blockIdx.y;
    const int tid = threadIdx.x;
    const float*  row  = A + ((size_t)r * NN + i) * NN;
    const float4* row4 = (const float4*)row;

    float s = 0.0f;
    #pragma unroll
    for (int p = 0; p < NN / (256 * 4); ++p) {     // 4 passes
        const float4 v = row4[p * 256 + tid];
        s += (v.x + v.y) + (v.z + v.w);
    }

    __shared__ float red[256];
    red[tid] = s;
    __syncthreads();
    for (int st = 128; st > 0; st >>= 1) {
        if (tid < st) red[tid] += red[tid + st];
        __syncthreads();
    }
    if (tid == 0) {
        S[r * NN + i]  = red[0];
        Dg[r * NN + i] = row[i];
    }
}

// ---------------------------------------------------------------------------
// K3: t_ii = tprev[i,i] * (sum_r w[r]*diag_r[i]) / (sum_r w[r]*rowsum_r[i])
// ---------------------------------------------------------------------------
__global__ __launch_bounds__(256) void k_tdiag(
    const float* __restrict__ w,
    const float* __restrict__ S,
    const float* __restrict__ Dg,
    const float* __restrict__ tprev,
    float*       __restrict__ tdiag)
{
    const int i = blockIdx.x * 256 + threadIdx.x;
    float num = 0.0f, den = 0.0f;
    #pragma unroll
    for (int r = 0; r < RR; ++r) {
        const float wr = w[r];
        num = fmaf(wr, Dg[r * NN + i], num);
        den = fmaf(wr, S [r * NN + i], den);
    }
    tdiag[i] = tprev[(size_t)i * NN + i] * num / den;
}

// ---------------------------------------------------------------------------
// K4: fused  out = LN( edges[node] + tdiag[i]*(edges @ W_v) + b_v )
// Block = 512 threads = 16 waves; block owns a 16-row band (full D=256 width),
// wave n computes the 16x16 tile at columns [16n, 16n+16) with 64 chained
// v_wmma_f32_16x16x4_f32 ops (K = 256). Epilogue: stage band in LDS, then
// wave n normalizes row n (wave32 shuffle reductions).
//
// FP32 WMMA lane layout (ISA 7.12.2):
//   A 16x4 : lanes 0-15 row M=lane {v0:K=0, v1:K=1}; lanes 16-31 {v0:K=2, v1:K=3}
//   B 4x16 : v0 {lanes 0-15: K=0, lanes 16-31: K=2}; v1 {K=1, K=3}, N=lane%16
//   C 16x16: v in 0..7 -> row v + 8*(lane/16), col = lane%16
// ---------------------------------------------------------------------------
__device__ inline float wave_sum32(float v) {
    #pragma unroll
    for (int off = 16; off > 0; off >>= 1) v += __shfl_xor(v, off, 32);
    return v;
}

__global__ __launch_bounds__(512) void k_gemm_ln(
    const float* __restrict__ edges,
    const float* __restrict__ W_v,
    const float* __restrict__ b_v,
    const float* __restrict__ tdiag,
    const int*   __restrict__ node_index,
    float*       __restrict__ out)
{
    const int m0   = blockIdx.x * 16;
    const int wave = threadIdx.x >> 5;   // 0..15 -> N-tile, then row in epilogue
    const int lane = threadIdx.x & 31;
    const int half = lane >> 4;          // 0|1
    const int lidx = lane & 15;
    const int n0   = wave * 16;

    v8f c = {};
    #pragma unroll 4
    for (int k = 0; k < DD; k += 4) {
        v2f a, b;
        const float* arow = edges + (size_t)(m0 + lidx) * DD + k + 2 * half;
        a[0] = arow[0];
        a[1] = arow[1];
        b[0] = W_v[(size_t)(k + 2 * half    ) * DD + n0 + lidx];
        b[1] = W_v[(size_t)(k + 2 * half + 1) * DD + n0 + lidx];
        c = __builtin_amdgcn_wmma_f32_16x16x4_f32(
                /*neg_a=*/false, a, /*neg_b=*/false, b,
                /*c_mod=*/(short)0, c, /*reuse_a=*/false, /*reuse_b=*/false);
    }

    __shared__ float tile[16][DD];       // 16 KB band: Y = edges @ W_v
    #pragma unroll
    for (int v = 0; v < 8; ++v)
        tile[v + 8 * half][n0 + lidx] = c[v];
    __syncthreads();

    // Epilogue: wave handles output row (m0 + wave).
    const int   row  = wave;
    const int   node = node_index[0];
    const float td   = tdiag[m0 + row];
    const float* enode = edges + (size_t)node * DD;

    float vals[8];
    float s = 0.0f;
    #pragma unroll
    for (int e = 0; e < 8; ++e) {
        const int col = lane + e * 32;
        const float v = enode[col] + td * tile[row][col] + b_v[col];
        vals[e] = v;
        s += v;
    }
    const float mean = wave_sum32(s) * (1.0f / DD);

    float vv = 0.0f;
    #pragma unroll
    for (int e = 0; e < 8; ++e) {
        const float d = vals[e] - mean;
        vv += d * d;
    }
    const float var = wave_sum32(vv) * (1.0f / DD);
    const float inv = rsqrtf(var + LN_EPS);

    float* orow = out + (size_t)(m0 + row) * DD;
    #pragma unroll
    for (int e = 0; e < 8; ++e) {
        const int col = lane + e * 32;
        orow[col] = (vals[e] - mean) * inv;
    }
}

// ---------------------------------------------------------------------------
extern "C" void kernel_launch(void* const* d_in, const int* in_sizes, int n_in,
                              void* d_out, int out_size, void* d_ws, size_t ws_size,
                              hipStream_t stream)
{
    const float* inputs_embeds = (const float*)d_in[0];
    const int*   token_index   = (const int*)  d_in[1];
    const int*   node_index    = (const int*)  d_in[2];
    const float* edges         = (const float*)d_in[3];
    const float* A             = (const float*)d_in[4];
    const float* rels          = (const float*)d_in[5];
    const float* wrel          = (const float*)d_in[6];
    const float* W_q           = (const float*)d_in[7];
    const float* b_q           = (const float*)d_in[8];
    const float* W_v           = (const float*)d_in[9];
    const float* b_v           = (const float*)d_in[10];
    const float* tprev         = (const float*)d_in[11];
    float*       out           = (float*)d_out;

    float* ws    = (float*)d_ws;
    float* w_buf = ws;                   // 8 floats (pad to 16)
    float* S     = ws + 16;              // R*N
    float* Dg    = S + RR * NN;          // R*N
    float* tdiag = Dg + RR * NN;         // N    -> total ~275 KB

    k_relpred<<<1, 256, 0, stream>>>(inputs_embeds, token_index, node_index,
                                     rels, wrel, W_q, b_q, w_buf);

    dim3 g2(NN, RR);
    k_rowsum<<<g2, 256, 0, stream>>>(A, S, Dg);

    k_tdiag<<<NN / 256, 256, 0, stream>>>(w_buf, S, Dg, tprev, tdiag);

    k_gemm_ln<<<NN / 16, 512, 0, stream>>>(edges, W_v, b_v, tdiag,
                                           node_index, out);
}